// unit_gcn_33346126086361
// MI455X (gfx1250) — compile-verified
//
#include <hip/hip_runtime.h>

// Problem constants (from reference)
#define N_    128
#define CIN_  64
#define COUT_ 64
#define T_    128
#define V_    25
#define S_    3
#define R_    8
#define TV_   (T_ * V_)          // 3200
#define BN_EPS 1e-5f
#define NCHUNK_ 100              // 409600 elems/channel / 4096
#define CHUNK_  4096

typedef __attribute__((ext_vector_type(2))) float v2f;
typedef __attribute__((ext_vector_type(8))) float v8f;

// ---------------------------------------------------------------------------
// CDNA5 fp32 WMMA: D(16x16) = A(16x4) * B(4x16) + C, wave32, fp32 throughout.
// A lane layout: a.{x,y} = A[lane&15][2*(lane>>4) + {0,1}]
// B lane layout: b.{x,y} = B[2*(lane>>4) + {0,1}][lane&15]
// C/D layout: vgpr i -> row = i + 8*(lane>>4), col = lane&15
// ---------------------------------------------------------------------------
__device__ __forceinline__ v8f wmma4(v2f a, v2f b, v8f c) {
#if defined(__gfx1250__)
  return __builtin_amdgcn_wmma_f32_16x16x4_f32(false, a, false, b, (short)0, c,
                                               false, false);
#else
  // non-gfx1250 pass stub (never executed on device)
  c[0] += a.x * b.x;
  return c;
#endif
}

// ---------------------------------------------------------------------------
__global__ void k_zero(float* __restrict__ p, long n) {
  long i = (long)blockIdx.x * blockDim.x + threadIdx.x;
  if (i < n) p[i] = 0.f;
}

// xmean[n,c,v] = mean_t x[n,c,t,v]   (fold mean-over-T before conv1 matmul)
__global__ void k_xmean(const float* __restrict__ x, float* __restrict__ xmean) {
  int i = blockIdx.x * blockDim.x + threadIdx.x;  // (n*CIN + c)*V + v
  if (i >= N_ * CIN_ * V_) return;
  int v = i % V_;
  int nc = i / V_;
  const float* p = x + (long)nc * TV_ + v;
  float s = 0.f;
  for (int t = 0; t < T_; ++t) s += p[t * V_];
  xmean[i] = s * (1.0f / T_);
}

// x3[n,o,tv] = sum_c w3[s][o][c] * x[n][c][tv] + b3[s][o]
// One wave per 16(M=o) x 16(N=tv) tile, K=CIN=64 in 16 wmma steps.
__global__ void k_gemm_x3(const float* __restrict__ x, const float* __restrict__ w3,
                          const float* __restrict__ b3, float* __restrict__ x3, int s) {
  int wave = blockIdx.x * (blockDim.x >> 5) + (threadIdx.x >> 5);
  int lane = threadIdx.x & 31;
  const int NTILE = TV_ / 16;   // 200
  const int MTILE = COUT_ / 16; // 4
  int nt = wave % NTILE;
  int rem = wave / NTILE;
  int mt = rem % MTILE;
  int n = rem / MTILE;
  if (n >= N_) return;  // wave-uniform

  int m0 = mt * 16, c0 = nt * 16;
  const float* W = w3 + (long)s * COUT_ * CIN_;
  const float* X = x + (long)n * CIN_ * TV_;
  int row = lane & 15;
  int kh = (lane >> 4) << 1;  // 0 or 2
  v8f acc = {0.f, 0.f, 0.f, 0.f, 0.f, 0.f, 0.f, 0.f};
  for (int k = 0; k < CIN_; k += 4) {
    v2f a, b;
    const float* wr = W + (m0 + row) * CIN_ + k + kh;
    a.x = wr[0];
    a.y = wr[1];
    b.x = X[(k + kh) * TV_ + c0 + row];
    b.y = X[(k + kh + 1) * TV_ + c0 + row];
    acc = wmma4(a, b, acc);
  }
  float* O = x3 + (long)n * COUT_ * TV_;
  const float* B3 = b3 + s * COUT_;
  int roff = (lane >> 4) * 8;
  for (int i = 0; i < 8; ++i) {
    int o = m0 + roff + i;
    O[(long)o * TV_ + c0 + row] = acc[i] + B3[o];
  }
}

// x1[n,r,v] = sum_c xmean[n,c,v] * w1[s][r][c] + b1[s][r]
__global__ void k_x1(const float* __restrict__ xmean, const float* __restrict__ w1,
                     const float* __restrict__ b1, float* __restrict__ x1, int s) {
  int i = blockIdx.x * blockDim.x + threadIdx.x;  // (n*R + r)*V + v
  if (i >= N_ * R_ * V_) return;
  int v = i % V_;
  int rem = i / V_;
  int r = rem % R_;
  int n = rem / R_;
  const float* W = w1 + (s * R_ + r) * CIN_;
  const float* XM = xmean + (long)n * CIN_ * V_ + v;
  float acc = b1[s * R_ + r];
  for (int c = 0; c < CIN_; ++c) acc += XM[c * V_] * W[c];
  x1[i] = acc;
}

// Dts[n,r,u,v] = tanh(x1[n,r,u] - x1[n,r,v])
__global__ void k_tanhdiff(const float* __restrict__ x1, float* __restrict__ Dts) {
  int i = blockIdx.x * blockDim.x + threadIdx.x;  // ((n*R+r)*V+u)*V+v
  if (i >= N_ * R_ * V_ * V_) return;
  int v = i % V_;
  int rem = i / V_;
  int u = rem % V_;
  int nr = rem / V_;
  Dts[i] = tanhf(x1[nr * V_ + u] - x1[nr * V_ + v]);
}

// A_S_s[n,o,u,v] = alpha*(sum_r Dts[n,r,u,v]*w4[s][o][r] + b4[s][o]) + PA[s][u][v]
// and ASout += A_S_s / S
__global__ void k_AS(const float* __restrict__ Dts, const float* __restrict__ w4,
                     const float* __restrict__ b4, const float* __restrict__ PA,
                     const float* __restrict__ alpha, float* __restrict__ ASs,
                     float* __restrict__ ASout, int s) {
  int i = blockIdx.x * blockDim.x + threadIdx.x;  // ((n*COUT+o)*V+u)*V+v
  if (i >= N_ * COUT_ * V_ * V_) return;
  int v = i % V_;
  int rem = i / V_;
  int u = rem % V_;
  rem /= V_;
  int o = rem % COUT_;
  int n = rem / COUT_;
  const float* W = w4 + (s * COUT_ + o) * R_;
  const float* D = Dts + ((long)n * R_ * V_ + u) * V_ + v;
  float acc = b4[s * COUT_ + o];
  for (int r = 0; r < R_; ++r) acc += D[(long)r * V_ * V_] * W[r];
  float as = acc * alpha[0] + PA[(s * V_ + u) * V_ + v];
  ASs[i] = as;
  ASout[i] += as * (1.0f / S_);
}

// y[n,c,t,u] += beta * sum_v A_S_s[n,c,u,v] * x3[n,c,t,v]
// One wave per (n,c, t-tile of 16, u-tile of 16); K=v padded 25->28.
__global__ void k_xs(const float* __restrict__ x3, const float* __restrict__ ASs,
                     const float* __restrict__ beta, float* __restrict__ yout) {
  int wave = blockIdx.x * (blockDim.x >> 5) + (threadIdx.x >> 5);
  int lane = threadIdx.x & 31;
  int ut = wave & 1;
  int rem = wave >> 1;
  int tt = rem & 7;
  rem >>= 3;
  int c = rem % COUT_;
  int n = rem / COUT_;
  if (n >= N_) return;

  int t0 = tt * 16, u0 = ut * 16;
  const float* X3 = x3 + ((long)n * COUT_ + c) * TV_;       // [t][v]
  const float* AS = ASs + ((long)n * COUT_ + c) * V_ * V_;  // [u][v]
  int row = lane & 15;
  int kh = (lane >> 4) << 1;
  int ucol = u0 + row;
  bool uin = ucol < V_;
  v8f acc = {0.f, 0.f, 0.f, 0.f, 0.f, 0.f, 0.f, 0.f};
  for (int kb = 0; kb < 28; kb += 4) {
    int k0 = kb + kh, k1 = k0 + 1;
    v2f a, b;
    a.x = (k0 < V_) ? X3[(t0 + row) * V_ + k0] : 0.f;
    a.y = (k1 < V_) ? X3[(t0 + row) * V_ + k1] : 0.f;
    b.x = (uin && k0 < V_) ? AS[ucol * V_ + k0] : 0.f;
    b.y = (uin && k1 < V_) ? AS[ucol * V_ + k1] : 0.f;
    acc = wmma4(a, b, acc);
  }
  float bt = beta[0];
  float* Y = yout + ((long)n * COUT_ + c) * TV_;
  int roff = (lane >> 4) * 8;
  if (uin) {
    for (int i = 0; i < 8; ++i) {
      int t = t0 + roff + i;
      Y[t * V_ + ucol] += bt * acc[i];
    }
  }
}

// m[n,t,v] = mean_c x3[n,c,t,v]
__global__ void k_m(const float* __restrict__ x3, float* __restrict__ m) {
  int i = blockIdx.x * blockDim.x + threadIdx.x;  // n*TV + tv
  if (i >= N_ * T_ * V_) return;
  int tv = i % TV_;
  int n = i / TV_;
  const float* p = x3 + (long)n * COUT_ * TV_ + tv;
  float s = 0.f;
  for (int c = 0; c < COUT_; ++c) s += p[(long)c * TV_];
  m[i] = s * (1.0f / COUT_);
}

// A_T_s[n,t,u,v] = tanh(m[n,t,u] - m[n,t,v]);  ATout += A_T_s / S
__global__ void k_AT(const float* __restrict__ m, float* __restrict__ ATs,
                     float* __restrict__ ATout) {
  long i = (long)blockIdx.x * blockDim.x + threadIdx.x;  // ((n*T+t)*V+u)*V+v
  if (i >= (long)N_ * T_ * V_ * V_) return;
  int v = (int)(i % V_);
  long rem = i / V_;
  int u = (int)(rem % V_);
  long nt = rem / V_;
  float at = tanhf(m[nt * V_ + u] - m[nt * V_ + v]);
  ATs[i] = at;
  ATout[i] += at * (1.0f / S_);
}

// y[n,c,t,u] += gamma * sum_v A_T_s[n,t,u,v] * x3[n,c,t,v]
// One wave per (n,t, c-tile of 16, u-tile of 16).
__global__ void k_xt(const float* __restrict__ x3, const float* __restrict__ ATs,
                     const float* __restrict__ gamma, float* __restrict__ yout) {
  int wave = blockIdx.x * (blockDim.x >> 5) + (threadIdx.x >> 5);
  int lane = threadIdx.x & 31;
  int ut = wave & 1;
  int rem = wave >> 1;
  int ct = rem & 3;
  rem >>= 2;
  int t = rem % T_;
  int n = rem / T_;
  if (n >= N_) return;

  int c0 = ct * 16, u0 = ut * 16;
  const float* X3 = x3 + (long)n * COUT_ * TV_ + t * V_;  // [c][v], c stride TV_
  const float* AT = ATs + ((long)n * T_ + t) * V_ * V_;   // [u][v]
  int row = lane & 15;
  int kh = (lane >> 4) << 1;
  int ucol = u0 + row;
  bool uin = ucol < V_;
  v8f acc = {0.f, 0.f, 0.f, 0.f, 0.f, 0.f, 0.f, 0.f};
  for (int kb = 0; kb < 28; kb += 4) {
    int k0 = kb + kh, k1 = k0 + 1;
    v2f a, b;
    a.x = (k0 < V_) ? X3[(long)(c0 + row) * TV_ + k0] : 0.f;
    a.y = (k1 < V_) ? X3[(long)(c0 + row) * TV_ + k1] : 0.f;
    b.x = (uin && k0 < V_) ? AT[ucol * V_ + k0] : 0.f;
    b.y = (uin && k1 < V_) ? AT[ucol * V_ + k1] : 0.f;
    acc = wmma4(a, b, acc);
  }
  float g = gamma[0];
  int roff = (lane >> 4) * 8;
  if (uin) {
    for (int i = 0; i < 8; ++i) {
      int c = c0 + roff + i;
      yout[(((long)n * COUT_ + c) * T_ + t) * V_ + ucol] += g * acc[i];
    }
  }
}

// BN stage 1: per (channel, chunk) partial sum / sumsq (deterministic, no atomics)
__global__ void k_bnpart(const float* __restrict__ yout, float* __restrict__ partials) {
  __shared__ float ssum[256];
  __shared__ float ssq[256];
  int c = blockIdx.x / NCHUNK_;
  int chunk = blockIdx.x % NCHUNK_;
  int tid = threadIdx.x;
  float s = 0.f, q = 0.f;
  for (int j = 0; j < CHUNK_ / 256; ++j) {
    int e = chunk * CHUNK_ + j * 256 + tid;  // flat (n,t,v) index
    int n = e / TV_;
    int rem = e % TV_;
    float val = yout[((long)n * COUT_ + c) * TV_ + rem];
    s += val;
    q += val * val;
  }
  ssum[tid] = s;
  ssq[tid] = q;
  __syncthreads();
  for (int k = 128; k > 0; k >>= 1) {
    if (tid < k) {
      ssum[tid] += ssum[tid + k];
      ssq[tid] += ssq[tid + k];
    }
    __syncthreads();
  }
  if (tid == 0) {
    partials[((long)c * NCHUNK_ + chunk) * 2 + 0] = ssum[0];
    partials[((long)c * NCHUNK_ + chunk) * 2 + 1] = ssq[0];
  }
}

// BN stage 2: reduce partials -> per-channel scale/shift
__global__ void k_bnstats(const float* __restrict__ partials, const float* __restrict__ bnw,
                          const float* __restrict__ bnb, float* __restrict__ stats) {
  __shared__ float ssum[128];
  __shared__ float ssq[128];
  int c = blockIdx.x;
  int tid = threadIdx.x;
  float s = 0.f, q = 0.f;
  if (tid < NCHUNK_) {
    s = partials[((long)c * NCHUNK_ + tid) * 2 + 0];
    q = partials[((long)c * NCHUNK_ + tid) * 2 + 1];
  }
  ssum[tid] = s;
  ssq[tid] = q;
  __syncthreads();
  for (int k = 64; k > 0; k >>= 1) {
    if (tid < k) {
      ssum[tid] += ssum[tid + k];
      ssq[tid] += ssq[tid + k];
    }
    __syncthreads();
  }
  if (tid == 0) {
    const float cnt = (float)(N_ * T_ * V_);
    float mu = ssum[0] / cnt;
    float var = ssq[0] / cnt - mu * mu;
    float scale = bnw[c] * rsqrtf(var + BN_EPS);
    stats[2 * c + 0] = scale;
    stats[2 * c + 1] = bnb[c] - mu * scale;
  }
}

// BN apply + residual + ReLU (in place over y region of d_out)
__global__ void k_bnapply(const float* __restrict__ x, const float* __restrict__ stats,
                          float* __restrict__ yout) {
  long i = (long)blockIdx.x * blockDim.x + threadIdx.x;
  if (i >= (long)N_ * COUT_ * TV_) return;
  int c = (int)((i / TV_) % COUT_);
  float val = yout[i] * stats[2 * c] + stats[2 * c + 1] + x[i];
  yout[i] = val > 0.f ? val : 0.f;
}

// ---------------------------------------------------------------------------
extern "C" void kernel_launch(void* const* d_in, const int* in_sizes, int n_in,
                              void* d_out, int out_size, void* d_ws, size_t ws_size,
                              hipStream_t stream) {
  (void)in_sizes; (void)n_in; (void)out_size; (void)ws_size;
  const float* x     = (const float*)d_in[0];
  const float* PA    = (const float*)d_in[1];
  const float* w1    = (const float*)d_in[2];
  const float* b1    = (const float*)d_in[3];
  const float* w3    = (const float*)d_in[4];
  const float* b3    = (const float*)d_in[5];
  const float* w4    = (const float*)d_in[6];
  const float* b4    = (const float*)d_in[7];
  const float* alpha = (const float*)d_in[8];
  const float* beta  = (const float*)d_in[9];
  const float* gamma = (const float*)d_in[10];
  const float* bnw   = (const float*)d_in[11];
  const float* bnb   = (const float*)d_in[12];

  float* yout  = (float*)d_out;                       // N*COUT*T*V
  float* ASout = yout + (long)N_ * COUT_ * TV_;       // N*COUT*V*V
  float* ATout = ASout + (long)N_ * COUT_ * V_ * V_;  // N*T*V*V

  // Workspace carve-up (~172 MB of fp32)
  float* wsf = (float*)d_ws;
  float* xmean = wsf;  wsf += (long)N_ * CIN_ * V_;
  float* x1    = wsf;  wsf += (long)N_ * R_ * V_;
  float* Dts   = wsf;  wsf += (long)N_ * R_ * V_ * V_;
  float* ASs   = wsf;  wsf += (long)N_ * COUT_ * V_ * V_;
  float* x3    = wsf;  wsf += (long)N_ * COUT_ * TV_;
  float* mbuf  = wsf;  wsf += (long)N_ * T_ * V_;
  float* ATs   = wsf;  wsf += (long)N_ * T_ * V_ * V_;
  float* partials = wsf;  wsf += (long)COUT_ * NCHUNK_ * 2;
  float* stats = wsf;  wsf += COUT_ * 2;

  // d_out accumulates over subsets -> must be zeroed every launch (determinism)
  long outN = (long)N_ * COUT_ * TV_ + (long)N_ * COUT_ * V_ * V_ + (long)N_ * T_ * V_ * V_;
  k_zero<<<(int)((outN + 255) / 256), 256, 0, stream>>>(yout, outN);

  k_xmean<<<(N_ * CIN_ * V_ + 255) / 256, 256, 0, stream>>>(x, xmean);

  for (int s = 0; s < S_; ++s) {
    // 1) x3 = conv3 (WMMA GEMM): 102,400 tile-waves, 8 waves/block
    k_gemm_x3<<<102400 / 8, 256, 0, stream>>>(x, w3, b3, x3, s);
    // 2) x1 = conv1 on T-mean
    k_x1<<<(N_ * R_ * V_ + 255) / 256, 256, 0, stream>>>(xmean, w1, b1, x1, s);
    // 3) pairwise tanh over R channels
    k_tanhdiff<<<(N_ * R_ * V_ * V_ + 255) / 256, 256, 0, stream>>>(x1, Dts);
    // 4) A_S build + running average into output
    k_AS<<<(N_ * COUT_ * V_ * V_ + 255) / 256, 256, 0, stream>>>(Dts, w4, b4, PA, alpha,
                                                                 ASs, ASout, s);
    // 5) spatial aggregation (WMMA): 131,072 tile-waves
    k_xs<<<131072 / 8, 256, 0, stream>>>(x3, ASs, beta, yout);
    // 6) channel mean
    k_m<<<(N_ * T_ * V_ + 255) / 256, 256, 0, stream>>>(x3, mbuf);
    // 7) A_T build + running average into output
    k_AT<<<(int)(((long)N_ * T_ * V_ * V_ + 255) / 256), 256, 0, stream>>>(mbuf, ATs, ATout);
    // 8) temporal-graph aggregation (WMMA): 131,072 tile-waves
    k_xt<<<131072 / 8, 256, 0, stream>>>(x3, ATs, gamma, yout);
  }

  // BatchNorm (two-stage deterministic reduction) + residual + ReLU
  k_bnpart<<<COUT_ * NCHUNK_, 256, 0, stream>>>(yout, partials);
  k_bnstats<<<COUT_, 128, 0, stream>>>(partials, bnw, bnb, stats);
  k_bnapply<<<(int)(((long)N_ * COUT_ * TV_ + 255) / 256), 256, 0, stream>>>(x, stats, yout);
}